// ConvQuadInterp3d_55327768707892
// MI455X (gfx1250) — compile-verified
//
#include <hip/hip_runtime.h>
#include <stdint.h>
#include <math.h>

// ConvQuadInterp3d on MI455X (gfx1250).
// Write-dominated stencil: stage 9 input rows per block into LDS with CDNA5
// async global->LDS loads (ASYNCcnt), compute 4 voxels/thread from a sliding
// register window, emit all outputs with non-temporal B128 stores so the
// 134MB of streaming output does not evict the 33.5MB input from L2.

typedef float f4 __attribute__((ext_vector_type(4)));
typedef int   v4i __attribute__((ext_vector_type(4)));

#ifndef __has_builtin
#define __has_builtin(x) 0
#endif

#define NB 4
#define ND 8
#define NH 512
#define NW 512

__device__ __forceinline__ int iclampi(int v, int lo, int hi) {
    return v < lo ? lo : (v > hi ? hi : v);
}

// CDNA5 async copy: global memory -> LDS, tracked by ASYNCcnt.
// Builtin signature (from hipcc diagnostic): param 0 is AS(1) v4i*, so pass
// typed pointers in the global (AS1) and LDS (AS3) address spaces.
__device__ __forceinline__ void async_ld_b128(const float* gsrc, float* ldst) {
#if __has_builtin(__builtin_amdgcn_global_load_async_to_lds_b128)
    __builtin_amdgcn_global_load_async_to_lds_b128(
        (__attribute__((address_space(1))) v4i*)gsrc,
        (__attribute__((address_space(3))) v4i*)ldst,
        0, 0);
#else
    unsigned lofs = (unsigned)(size_t)(__attribute__((address_space(3))) void*)ldst;
    asm volatile("global_load_async_to_lds_b128 %0, %1, off"
                 :: "v"(lofs), "v"(gsrc)
                 : "memory");
#endif
}

__device__ __forceinline__ void wait_async0() {
#if __has_builtin(__builtin_amdgcn_s_wait_asynccnt)
    __builtin_amdgcn_s_wait_asynccnt(0);
#else
    asm volatile("s_wait_asynccnt 0" ::: "memory");
#endif
}

// Streaming outputs: non-temporal so they don't pollute L2 (input reuse lives there).
__device__ __forceinline__ void nt_store4(float* p, f4 v) {
    __builtin_nontemporal_store(v, (f4*)p);
}

extern "C" __global__ __launch_bounds__(128)
void conv_quad_interp3d_kernel(const float* __restrict__ x, float* __restrict__ out) {
    __shared__ float lds[9 * NW];   // 18 KB of 320 KB/WGP

    const int tid = threadIdx.x;    // 0..127, each owns 4 consecutive w
    const int h = blockIdx.x;
    const int d = blockIdx.y;
    const int b = blockIdx.z;
    const int w0 = tid * 4;

    // ---- Stage 9 replicate-clamped rows (3 depth x 3 height) into LDS ----
    #pragma unroll
    for (int dz = 0; dz < 3; ++dz) {
        const int gd = iclampi(d + dz - 1, 0, ND - 1);
        #pragma unroll
        for (int hy = 0; hy < 3; ++hy) {
            const int gh = iclampi(h + hy - 1, 0, NH - 1);
            const float* src = x + (((size_t)(b * ND + gd) * NH + gh) * NW) + w0;
            float* dst = &lds[(dz * 3 + hy) * NW + w0];
            async_ld_b128(src, dst);
        }
    }
    wait_async0();        // this wave's async LDS writes landed
    __syncthreads();      // all waves' writes visible

    // ---- 6-wide sliding window per row: [w-1, w0..w0+3, w+4] (clamped) ----
    float win[3][3][6];
    #pragma unroll
    for (int dz = 0; dz < 3; ++dz) {
        #pragma unroll
        for (int hy = 0; hy < 3; ++hy) {
            const float* row = &lds[(dz * 3 + hy) * NW];
            const f4 m = *(const f4*)(row + w0);            // ds_load_b128
            win[dz][hy][0] = row[w0 ? (w0 - 1) : 0];
            win[dz][hy][1] = m.x;
            win[dz][hy][2] = m.y;
            win[dz][hy][3] = m.z;
            win[dz][hy][4] = m.w;
            win[dz][hy][5] = row[(w0 + 4 < NW) ? (w0 + 4) : (NW - 1)];
        }
    }

    f4 yv4, c0v, c1v, c2v;

    #pragma unroll
    for (int j = 0; j < 4; ++j) {
        // W = w-1, C = w, E = w+1 for each (depth dz, height hy) row
        const float W11 = win[1][1][j], C11 = win[1][1][j + 1], E11 = win[1][1][j + 2];
        const float W10 = win[1][0][j], C10 = win[1][0][j + 1], E10 = win[1][0][j + 2];
        const float W12 = win[1][2][j], C12 = win[1][2][j + 1], E12 = win[1][2][j + 2];
        const float W01 = win[0][1][j], C01 = win[0][1][j + 1], E01 = win[0][1][j + 2];
        const float W21 = win[2][1][j], C21 = win[2][1][j + 1], E21 = win[2][1][j + 2];
        const float C00 = win[0][0][j + 1], C02 = win[0][2][j + 1];
        const float C20 = win[2][0][j + 1], C22 = win[2][2][j + 1];

        const float c = C11;
        const float bx = 0.5f * (E11 - W11);
        const float by = 0.5f * (C12 - C10);
        const float bs = 0.5f * (C21 - C01);
        const float dxx = E11 + W11 - 2.0f * c;
        const float dyy = C12 + C10 - 2.0f * c;
        const float dss = C21 + C01 - 2.0f * c;
        const float dxy = W10 - E10 - W12 + E12;
        const float dys = C00 - C02 - C20 + C22;
        const float dxs = W01 - E01 - W21 + E21;

        // Symmetric Hessian [[a,d,e],[d,b,f],[e,f,c]]: shared adjugate entries
        const float aM = dxx, bM = dyy, cM = dss;
        const float dM = 0.25f * dxy, fM = 0.25f * dys, eM = 0.25f * dxs;
        const float A00 = bM * cM - fM * fM;
        const float A01 = eM * fM - dM * cM;
        const float A02 = dM * fM - eM * bM;
        const float A11 = aM * cM - eM * eM;
        const float A12 = eM * dM - aM * fM;
        const float A22 = aM * bM - dM * dM;
        const float det = aM * A00 + dM * A01 + eM * A02;

        const bool okv = fabsf(det) > 1e-7f;
        const float inv = 1.0f / (okv ? det : 1.0f);   // one divide, three muls
        const float sx = (A00 * bx + A01 * by + A02 * bs) * inv;
        const float sy = (A01 * bx + A11 * by + A12 * bs) * inv;
        const float ss = (A02 * bx + A12 * by + A22 * bs) * inv;

        // Strict 3x3x3 NMS. Clamp-replicated values duplicate in-window
        // values, so this equals the reference's -inf-padded (clipped) max.
        float mx = c;
        #pragma unroll
        for (int dz = 0; dz < 3; ++dz)
            #pragma unroll
            for (int hy = 0; hy < 3; ++hy)
                #pragma unroll
                for (int k = 0; k < 3; ++k)
                    mx = fmaxf(mx, win[dz][hy][j + k]);
        const bool nm = (c == mx) && okv;

        float dxv = nm ? -sx : 0.0f;
        float dyv = nm ? -sy : 0.0f;
        float dsv = nm ? -ss : 0.0f;
        const bool over =
            fmaxf(fmaxf(fabsf(dxv), fabsf(dyv)), fabsf(dsv)) > 0.7f;
        if (over) { dxv = 0.0f; dyv = 0.0f; dsv = 0.0f; }

        yv4[j] = c + 0.5f * (bx * dxv + by * dyv + bs * dsv) + (nm ? 10.0f : 0.0f);
        c0v[j] = dsv + (float)d;          // scale/depth channel
        c1v[j] = dxv + (float)(w0 + j);   // x/width channel
        c2v[j] = dyv + (float)h;          // y/height channel
    }

    // ---- Outputs: coords (B,C=1,3,D,H,W) then y_max (B,C=1,D,H,W) ----
    const size_t planeHW = (size_t)NH * NW;
    const size_t rowOff = (size_t)h * NW + (size_t)w0;
    const size_t coordsVol = (size_t)NB * 3 * ND * planeHW;
    const size_t volBase = (size_t)(b * ND + d) * planeHW + rowOff;

    nt_store4(out + ((size_t)(b * 3 + 0) * ND + d) * planeHW + rowOff, c0v);
    nt_store4(out + ((size_t)(b * 3 + 1) * ND + d) * planeHW + rowOff, c1v);
    nt_store4(out + ((size_t)(b * 3 + 2) * ND + d) * planeHW + rowOff, c2v);
    nt_store4(out + coordsVol + volBase, yv4);
}

extern "C" void kernel_launch(void* const* d_in, const int* in_sizes, int n_in,
                              void* d_out, int out_size, void* d_ws, size_t ws_size,
                              hipStream_t stream) {
    (void)in_sizes; (void)n_in; (void)out_size; (void)d_ws; (void)ws_size;
    const float* x = (const float*)d_in[0];
    float* out = (float*)d_out;

    dim3 grid(NH, ND, NB);   // one (b,d,h) row of 512 voxels per block
    dim3 block(128);         // 4 wave32 waves; 4 voxels per thread
    conv_quad_interp3d_kernel<<<grid, block, 0, stream>>>(x, out);
}